// Attn_36524401885211
// MI455X (gfx1250) — compile-verified
//
#include <hip/hip_runtime.h>
#include <hip/hip_bf16.h>

typedef float v2f __attribute__((ext_vector_type(2)));
typedef float v4f __attribute__((ext_vector_type(4)));
typedef float v8f __attribute__((ext_vector_type(8)));

#define B_ 32
#define L_ 2048
#define H_ 2048

// ---------------------------------------------------------------------------
// Kernel 1: u[b,h] = sum_g hidden[b,g] * W[g,h]   (M=32, N=2048, K=2048)
// V_WMMA_F32_16X16X4_F32, one wave per 16x16 output tile, 8 waves/block
// A (16x4 f32): lane m=lane&15; VGPR0 holds K = k + 2*(lane>=16), VGPR1 = +1
// B (4x16 f32): lane n=lane&15; VGPR0 holds row K = k + 2*(lane>=16), VGPR1 = +1
// C/D (16x16 f32): VGPR r -> M = r + 8*(lane>=16), N = lane&15
// ---------------------------------------------------------------------------
__global__ __launch_bounds__(256)
void proj_wmma_kernel(const float* __restrict__ hidden,
                      const float* __restrict__ W,
                      float* __restrict__ u) {
    const int lane  = threadIdx.x & 31;
    const int wave  = threadIdx.x >> 5;
    const int ntile = blockIdx.x * 8 + wave;   // 128 N tiles
    const int mtile = blockIdx.y;              // 2 M tiles
    const int n0 = ntile * 16;
    const int m0 = mtile * 16;

    const int mrow  = m0 + (lane & 15);
    const int ncol  = n0 + (lane & 15);
    const int khalf = (lane >> 4) * 2;         // 0 or 2

    const float* hrow = hidden + (size_t)mrow * H_;

    v8f acc = {};
#pragma unroll 4
    for (int k = 0; k < H_; k += 4) {
        const int kk = k + khalf;
        // A fragment: two consecutive K values of this lane's hidden row
        v2f a = *(const v2f*)(hrow + kk);                  // 8B aligned (kk even)
        // B fragment: two consecutive K rows of W at column ncol
        v2f b;
        b.x = W[(size_t)kk * H_ + ncol];
        b.y = W[(size_t)(kk + 1) * H_ + ncol];
        acc = __builtin_amdgcn_wmma_f32_16x16x4_f32(
            /*neg_a=*/false, a, /*neg_b=*/false, b,
            /*c_mod=*/(short)0, acc, /*reuse_a=*/false, /*reuse_b=*/false);
    }

    const int moff = (lane >> 4) * 8;
#pragma unroll
    for (int r = 0; r < 8; ++r) {
        u[(size_t)(m0 + r + moff) * H_ + ncol] = acc[r];
    }
}

// ---------------------------------------------------------------------------
// Kernel 2: c[b] = dot(bias, hidden[b,:])   (one wave per batch)
// ---------------------------------------------------------------------------
__global__ __launch_bounds__(32)
void bias_dot_kernel(const float* __restrict__ bias,
                     const float* __restrict__ hidden,
                     float* __restrict__ c) {
    const int b = blockIdx.x;
    const int lane = threadIdx.x;
    float s = 0.f;
    for (int g = lane; g < H_; g += 32)
        s += bias[g] * hidden[(size_t)b * H_ + g];
#pragma unroll
    for (int m = 16; m; m >>= 1) s += __shfl_xor(s, m, 32);
    if (lane == 0) c[b] = s;
}

// ---------------------------------------------------------------------------
// Kernel 3: energies[b,l] = dot(enc[b,l,:], u[b,:]) + c[b]
// Streams 512 MiB of enc exactly once -> bandwidth-bound (~22 us @ 23.3 TB/s).
// u[b] staged in LDS; enc read with non-temporal b128 loads; one wave per row.
// grid = (L/64, B), block = 256 (8 waves, 8 rows each)
// ---------------------------------------------------------------------------
__global__ __launch_bounds__(256)
void energies_kernel(const float* __restrict__ enc,
                     const float* __restrict__ u,
                     const float* __restrict__ c,
                     float* __restrict__ energies) {
    __shared__ v4f u_s[H_ / 4];
    const int b    = blockIdx.y;
    const int lane = threadIdx.x & 31;
    const int wave = threadIdx.x >> 5;

    const v4f* usrc = (const v4f*)(u + (size_t)b * H_);
    for (int i = threadIdx.x; i < H_ / 4; i += 256) u_s[i] = usrc[i];
    __syncthreads();

    const float cb = c[b];
    const int rowbase = blockIdx.x * 64 + wave * 8;

    for (int rr = 0; rr < 8; ++rr) {
        const int l = rowbase + rr;
        const v4f* row = (const v4f*)(enc + ((size_t)b * L_ + (size_t)l) * H_);
        float s = 0.f;
#pragma unroll
        for (int i = 0; i < H_ / 128; ++i) {           // 16 iterations
            const int j = lane + 32 * i;
            v4f e = __builtin_nontemporal_load(&row[j]);   // streamed once: NT hint
            v4f uu = u_s[j];
            s += e.x * uu.x + e.y * uu.y + e.z * uu.z + e.w * uu.w;
        }
#pragma unroll
        for (int m = 16; m; m >>= 1) s += __shfl_xor(s, m, 32);
        if (lane == 0) energies[(size_t)b * L_ + l] = s + cb;
    }
}

// ---------------------------------------------------------------------------
// Kernel 4: softmax over L per batch. grid = B, block = 256.
// ---------------------------------------------------------------------------
__global__ __launch_bounds__(256)
void softmax_kernel(const float* __restrict__ energies,
                    float* __restrict__ out) {
    __shared__ float e_s[L_];
    __shared__ float red[8];
    const int b    = blockIdx.x;
    const int tid  = threadIdx.x;
    const int lane = tid & 31;
    const int wave = tid >> 5;

    float lmax = -INFINITY;
    for (int i = tid; i < L_; i += 256) {
        float v = energies[(size_t)b * L_ + i];
        e_s[i] = v;
        lmax = fmaxf(lmax, v);
    }
#pragma unroll
    for (int m = 16; m; m >>= 1) lmax = fmaxf(lmax, __shfl_xor(lmax, m, 32));
    if (lane == 0) red[wave] = lmax;
    __syncthreads();
    float bmax = red[0];
#pragma unroll
    for (int w = 1; w < 8; ++w) bmax = fmaxf(bmax, red[w]);
    __syncthreads();

    float lsum = 0.f;
    for (int i = tid; i < L_; i += 256) {
        float v = __expf(e_s[i] - bmax);
        e_s[i] = v;
        lsum += v;
    }
#pragma unroll
    for (int m = 16; m; m >>= 1) lsum += __shfl_xor(lsum, m, 32);
    if (lane == 0) red[wave] = lsum;
    __syncthreads();
    float bsum = 0.f;
#pragma unroll
    for (int w = 0; w < 8; ++w) bsum += red[w];
    const float inv = 1.0f / bsum;
    for (int i = tid; i < L_; i += 256)
        out[(size_t)b * L_ + i] = e_s[i] * inv;
}

// ---------------------------------------------------------------------------
extern "C" void kernel_launch(void* const* d_in, const int* in_sizes, int n_in,
                              void* d_out, int out_size, void* d_ws, size_t ws_size,
                              hipStream_t stream) {
    const float* hidden = (const float*)d_in[0];  // [B,1,H]
    const float* enc    = (const float*)d_in[1];  // [B,L,H]
    const float* W      = (const float*)d_in[2];  // [H,H]
    const float* bias   = (const float*)d_in[3];  // [H]
    float* out = (float*)d_out;                   // [B,L,1]

    float* ws       = (float*)d_ws;
    float* u        = ws;                         // B*H floats
    float* cvec     = u + (size_t)B_ * H_;        // B floats (padded to 64)
    float* energies = cvec + 64;                  // B*L floats

    // 1) u = hidden @ W  (WMMA f32 16x16x4)
    proj_wmma_kernel<<<dim3(H_ / 16 / 8, B_ / 16), 256, 0, stream>>>(hidden, W, u);
    // 2) c[b] = bias . hidden[b]
    bias_dot_kernel<<<B_, 32, 0, stream>>>(bias, hidden, cvec);
    // 3) energies = batched enc @ u + c   (bandwidth-bound streaming pass)
    energies_kernel<<<dim3(L_ / 64, B_), 256, 0, stream>>>(enc, u, cvec, energies);
    // 4) softmax over L
    softmax_kernel<<<B_, 256, 0, stream>>>(energies, out);
}